// UnetConv3_deformable_81080392614506
// MI455X (gfx1250) — compile-verified
//
#include <hip/hip_runtime.h>
#include <hip/hip_bf16.h>

// CDNA5 / gfx1250 deformable UNet conv block, fp32 end-to-end using
// V_WMMA_F32_16X16X4_F32 implicit GEMMs, 4-way N-blocked per wave.

typedef __attribute__((ext_vector_type(2))) float v2f;
typedef __attribute__((ext_vector_type(8))) float v8f;

#define S_VOX   32768          // 32*32*32 voxels
#define DIM     32
#define PADC    1
#define NTILE   64             // voxels per block (4 x 16-wide WMMA tiles)

// ---------------------------------------------------------------------------
// Generic implicit-GEMM 3x3x3 conv, pad 1.
// out[m][s] = bias[m] + sum_kk wgt[m*Ktot+kk] * im2col(kk, s),
// kk = c*27 + (kd*9 + kh*3 + kw).
// block = 128 threads (4 waves). blockIdx.x = N-tile (64 voxels),
// blockIdx.y = M-block (64 rows). Each wave owns one 16-row M-tile and
// 4 accumulators spanning the 64 voxels (A fragment reused 4x).
// ---------------------------------------------------------------------------
__global__ void conv3d_wmma_f32(const float* __restrict__ in,
                                const float* __restrict__ wgt,
                                const float* __restrict__ bias,
                                float* __restrict__ out,
                                int Cin, int Ocnt) {
    __shared__ float Bs[32 * NTILE];            // K=32 x N=64 panel (8 KB)

    const int lane = threadIdx.x & 31;
    const int wave = threadIdx.x >> 5;
    const int nl   = lane & 15;
    const int half = lane >> 4;                 // 0: K pair {0,1}, 1: {2,3}
    const int Ktot = Cin * 27;
    const int Mbase = blockIdx.y * 64 + wave * 16;
    const int Mrow  = Mbase + nl;               // A row for this lane
    const bool am   = (Mrow < Ocnt);

    v8f acc[4];
#pragma unroll
    for (int t = 0; t < 4; ++t)
        acc[t] = (v8f){0.f, 0.f, 0.f, 0.f, 0.f, 0.f, 0.f, 0.f};

    // B-staging role: each thread owns one K-row, sweeps 16 columns
    const int srow = threadIdx.x >> 2;          // 0..31
    const int scol = (threadIdx.x & 3) * 16;    // column base

    for (int k0 = 0; k0 < Ktot; k0 += 32) {
        if (am) __builtin_prefetch(&wgt[Mrow * Ktot + k0], 0, 1);

        // decode this thread's K-row ONCE, then sweep columns
        {
            int kk = k0 + srow;
            int cc = kk / 27, tt = kk % 27;
            int kd = tt / 9, kh = (tt % 9) / 3, kw = tt % 3;
            const float* base = in + cc * S_VOX;
#pragma unroll 4
            for (int q = 0; q < 16; ++q) {
                int nn = scol + q;
                int sv = blockIdx.x * NTILE + nn;
                int d = sv >> 10, h = (sv >> 5) & 31, w = sv & 31;
                int z = d + kd - PADC, y = h + kh - PADC, x = w + kw - PADC;
                float v = 0.f;
                if ((unsigned)z < 32u && (unsigned)y < 32u && (unsigned)x < 32u)
                    v = base[(z * DIM + y) * DIM + x];
                Bs[srow * NTILE + nn] = v;
            }
        }
        __syncthreads();

#pragma unroll
        for (int j = 0; j < 8; ++j) {
            int kk = k0 + j * 4 + half * 2;
            v2f a;
            a.x = am ? wgt[Mrow * Ktot + kk]     : 0.f;
            a.y = am ? wgt[Mrow * Ktot + kk + 1] : 0.f;
            int r0 = j * 4 + half * 2;
#pragma unroll
            for (int t = 0; t < 4; ++t) {
                v2f b;
                b.x = Bs[r0 * NTILE + t * 16 + nl];
                b.y = Bs[(r0 + 1) * NTILE + t * 16 + nl];
                acc[t] = __builtin_amdgcn_wmma_f32_16x16x4_f32(
                    false, a, false, b, (short)0, acc[t], false, false);
            }
        }
        __syncthreads();
    }

#pragma unroll
    for (int t = 0; t < 4; ++t) {
        int svox = blockIdx.x * NTILE + t * 16 + nl;
#pragma unroll
        for (int j = 0; j < 8; ++j) {
            int m = Mbase + j + half * 8;
            if (m < Ocnt)
                out[m * S_VOX + svox] = acc[t][j] + (bias ? bias[m] : 0.f);
        }
    }
}

// ---------------------------------------------------------------------------
// Deformable conv: trilinear-sampled im2col panel -> WMMA GEMM.
// offs channel layout: axis*27 + tap (axis 0=d, 1=h, 2=w).
// p = (coord+1) + (k-1) + off, clipped to [0,33] (padded grid),
// i1 = min(i0+1, 33); 8-corner blend on zero-padded input.
// block = 128 threads; all 64 output channels (4 waves); Ktot = 32*27.
// ---------------------------------------------------------------------------
__global__ void deform_conv3d_wmma_f32(const float* __restrict__ in,
                                       const float* __restrict__ offs,
                                       const float* __restrict__ wgt,
                                       float* __restrict__ out) {
    __shared__ int   Pi[27 * NTILE * 3];        // floor idx (padded coords)
    __shared__ float Pf[27 * NTILE * 3];        // fracs
    __shared__ float Bs[32 * NTILE];

    const int lane = threadIdx.x & 31;
    const int wave = threadIdx.x >> 5;
    const int nl   = lane & 15;
    const int half = lane >> 4;
    const int Ktot = 32 * 27;                   // Cin = 32
    const int Mbase = wave * 16;
    const int Mrow  = Mbase + nl;

    // stage positions once per block: 27 taps x 64 voxels
    for (int e = threadIdx.x; e < 27 * NTILE; e += 128) {
        int tap = e / NTILE, nn = e & (NTILE - 1);
        int sv = blockIdx.x * NTILE + nn;
        int d = sv >> 10, h = (sv >> 5) & 31, w = sv & 31;
        int kd = tap / 9, kh = (tap % 9) / 3, kw = tap % 3;
        float od = offs[(0 * 27 + tap) * S_VOX + sv];
        float oh = offs[(1 * 27 + tap) * S_VOX + sv];
        float ow = offs[(2 * 27 + tap) * S_VOX + sv];
        float pd = fminf(fmaxf((float)(d + kd) + od, 0.f), 33.f); // d+1+(kd-1)
        float ph = fminf(fmaxf((float)(h + kh) + oh, 0.f), 33.f);
        float pw = fminf(fmaxf((float)(w + kw) + ow, 0.f), 33.f);
        float qd = floorf(pd), qh = floorf(ph), qw = floorf(pw);
        Pi[e * 3 + 0] = (int)qd; Pi[e * 3 + 1] = (int)qh; Pi[e * 3 + 2] = (int)qw;
        Pf[e * 3 + 0] = pd - qd; Pf[e * 3 + 1] = ph - qh; Pf[e * 3 + 2] = pw - qw;
    }
    __syncthreads();

    v8f acc[4];
#pragma unroll
    for (int t = 0; t < 4; ++t)
        acc[t] = (v8f){0.f, 0.f, 0.f, 0.f, 0.f, 0.f, 0.f, 0.f};

    const int srow = threadIdx.x >> 2;          // 0..31
    const int scol = (threadIdx.x & 3) * 16;

    for (int k0 = 0; k0 < Ktot; k0 += 32) {
        // decode K-row once: channel + tap
        {
            int kk = k0 + srow;
            int cc = kk / 27, tap = kk % 27;
            const float* inc = in + cc * S_VOX;
            for (int q = 0; q < 16; ++q) {
                int nn  = scol + q;
                int idx = (tap * NTILE + nn) * 3;
                int d0 = Pi[idx], h0 = Pi[idx + 1], w0 = Pi[idx + 2];
                float fd = Pf[idx], fh = Pf[idx + 1], fw = Pf[idx + 2];
                int d1 = min(d0 + 1, 33), h1 = min(h0 + 1, 33), w1 = min(w0 + 1, 33);
                auto samp = [&](int z, int y, int x) -> float {
                    z -= PADC; y -= PADC; x -= PADC;   // padded -> raw coords
                    return ((unsigned)z < 32u && (unsigned)y < 32u && (unsigned)x < 32u)
                               ? inc[(z * DIM + y) * DIM + x] : 0.f;
                };
                float gd = 1.f - fd, gh = 1.f - fh, gw = 1.f - fw;
                float v = samp(d0, h0, w0) * gd * gh * gw
                        + samp(d0, h0, w1) * gd * gh * fw
                        + samp(d0, h1, w0) * gd * fh * gw
                        + samp(d0, h1, w1) * gd * fh * fw
                        + samp(d1, h0, w0) * fd * gh * gw
                        + samp(d1, h0, w1) * fd * gh * fw
                        + samp(d1, h1, w0) * fd * fh * gw
                        + samp(d1, h1, w1) * fd * fh * fw;
                Bs[srow * NTILE + nn] = v;
            }
        }
        __syncthreads();

#pragma unroll
        for (int j = 0; j < 8; ++j) {
            int kk = k0 + j * 4 + half * 2;
            v2f a;
            a.x = wgt[Mrow * Ktot + kk];
            a.y = wgt[Mrow * Ktot + kk + 1];
            int r0 = j * 4 + half * 2;
#pragma unroll
            for (int t = 0; t < 4; ++t) {
                v2f b;
                b.x = Bs[r0 * NTILE + t * 16 + nl];
                b.y = Bs[(r0 + 1) * NTILE + t * 16 + nl];
                acc[t] = __builtin_amdgcn_wmma_f32_16x16x4_f32(
                    false, a, false, b, (short)0, acc[t], false, false);
            }
        }
        __syncthreads();
    }

#pragma unroll
    for (int t = 0; t < 4; ++t) {
        int svox = blockIdx.x * NTILE + t * 16 + nl;
#pragma unroll
        for (int j = 0; j < 8; ++j) {
            int m = Mbase + j + half * 8;
            out[m * S_VOX + svox] = acc[t][j];
        }
    }
}

// ---------------------------------------------------------------------------
// Batchnorm: per-channel sum / sumsq (one block per channel)
// ---------------------------------------------------------------------------
__global__ void bn_stats(const float* __restrict__ x, float* __restrict__ sums, int C) {
    __shared__ float rs[256], rq[256];
    int c = blockIdx.x;
    float s = 0.f, q = 0.f;
    for (int i = threadIdx.x; i < S_VOX; i += blockDim.x) {
        float v = x[c * S_VOX + i];
        s += v; q += v * v;
    }
    rs[threadIdx.x] = s; rq[threadIdx.x] = q;
    __syncthreads();
    for (int off = blockDim.x >> 1; off > 0; off >>= 1) {
        if (threadIdx.x < off) {
            rs[threadIdx.x] += rs[threadIdx.x + off];
            rq[threadIdx.x] += rq[threadIdx.x + off];
        }
        __syncthreads();
    }
    if (threadIdx.x == 0) { sums[c] = rs[0]; sums[C + c] = rq[0]; }
}

// normalize + ReLU (may run in place: y == x)
__global__ void bn_relu(const float* __restrict__ x, const float* __restrict__ sums,
                        const float* __restrict__ gamma, const float* __restrict__ beta,
                        float* __restrict__ y, int C) {
    int idx = blockIdx.x * blockDim.x + threadIdx.x;
    if (idx >= C * S_VOX) return;
    int c = idx >> 15;
    float inv  = 1.f / (float)S_VOX;
    float mean = sums[c] * inv;
    float var  = sums[C + c] * inv - mean * mean;
    float v = gamma[c] * (x[idx] - mean) * rsqrtf(var + 1e-5f) + beta[c];
    y[idx] = fmaxf(v, 0.f);
}

// ---------------------------------------------------------------------------
extern "C" void kernel_launch(void* const* d_in, const int* in_sizes, int n_in,
                              void* d_out, int out_size, void* d_ws, size_t ws_size,
                              hipStream_t stream) {
    const float* inputs = (const float*)d_in[0];
    const float* w_off  = (const float*)d_in[1];
    const float* b_off  = (const float*)d_in[2];
    const float* w_def  = (const float*)d_in[3];
    const float* gamma1 = (const float*)d_in[4];
    const float* beta1  = (const float*)d_in[5];
    const float* w2     = (const float*)d_in[6];
    const float* b2     = (const float*)d_in[7];
    const float* gamma2 = (const float*)d_in[8];
    const float* beta2  = (const float*)d_in[9];

    float* y_out    = (float*)d_out;            // (64, 32768)
    float* offs_out = y_out + 64 * S_VOX;       // (81, 32768)
    float* y_pre    = (float*)d_ws;             // (64, 32768) scratch
    float* stats    = y_pre + 64 * S_VOX;       // 2*64 floats

    const dim3 blk(128);
    const int NT = S_VOX / NTILE;               // 512 N-tiles

    // 1) offsets = conv3d(inputs, w_off) + b_off     (M=81 -> 2 M-blocks)
    conv3d_wmma_f32<<<dim3(NT, 2), blk, 0, stream>>>(inputs, w_off, b_off,
                                                     offs_out, 32, 81);
    // 2) y_pre = deform_conv3d(inputs, offsets, w_def)
    deform_conv3d_wmma_f32<<<dim3(NT, 1), blk, 0, stream>>>(inputs, offs_out,
                                                            w_def, y_pre);
    // 3) y1 = relu(bn(y_pre))  (in place)
    bn_stats<<<64, 256, 0, stream>>>(y_pre, stats, 64);
    bn_relu<<<(64 * S_VOX + 255) / 256, 256, 0, stream>>>(y_pre, stats, gamma1,
                                                          beta1, y_pre, 64);
    // 4) y2_pre = conv3d(y1, w2) + b2  -> write into d_out y-slot
    conv3d_wmma_f32<<<dim3(NT, 1), blk, 0, stream>>>(y_pre, w2, b2,
                                                     y_out, 64, 64);
    // 5) y = relu(bn(y2_pre))  (in place in d_out)
    bn_stats<<<64, 256, 0, stream>>>(y_out, stats, 64);
    bn_relu<<<(64 * S_VOX + 255) / 256, 256, 0, stream>>>(y_out, stats, gamma2,
                                                          beta2, y_out, 64);
}